// ConcatVolume_85048942395511
// MI455X (gfx1250) — compile-verified
//
#include <hip/hip_runtime.h>

// ConcatVolume: out(B=2, 2C=64, D=48, H=128, W=240) fp32 from left/right(B,C,H,W).
//   c <  C : out[b,c,d,h,w] = left [b,c,h,w]     * (w >= d)
//   c >= C : out[b,c,d,h,w] = right[b,c-C,h,w-d] * (w >= d)
//
// Store-bound: 755 MB written once -> ~32 us floor @ 23.3 TB/s. Inputs
// (15.7 MB) are L2-resident (192 MB L2). Design:
//  - single fused kernel, block-uniform SALU-only index math (no divisions)
//  - thread = (w4, dsub); fully unrolled d = dsub + 4i loop -> store
//    addresses are compile-time immediate offsets (max 5.4 MB < 24-bit)
//  - NT-hinted b128 stores (gfx1250 TH=NT): write-once stream, keep L2 for
//    the inputs
//  - left: 1 b128 load per 12 stores; mask-free fast path for lanes with
//    w0 >= 48 (never masked since d <= 47)
//  - right: rolling 2-register window (d+=4 shifts source exactly one
//    float4): 1 b128 load + 1 NT b128 store per iteration, dsub-templated
//    wave-uniform select; clamped loads never leak (mask == in-range)

typedef float v4f __attribute__((ext_vector_type(4)));

namespace cv {
constexpr int kB  = 2;
constexpr int kC  = 32;
constexpr int kH  = 128;
constexpr int kW  = 240;
constexpr int kD  = 48;
constexpr int kW4 = kW / 4;        // 60 float4s per row
constexpr int kHW = kH * kW;       // 30720 floats: out stride per d step
constexpr int kDI = kD / 4;        // 12 iterations per thread
}  // namespace cv

// Right-half inner loop, specialized on dsub (wave-uniform).
template <int DS>
__device__ __forceinline__ void right_loop(const float* __restrict__ rrow,
                                           float* __restrict__ o, int w0,
                                           v4f cur, v4f prev) {
  using namespace cv;
#pragma unroll
  for (int i = 0; i < kDI; ++i) {
    const int d = DS + 4 * i;
    // Window R[w0-d .. w0+3-d]; cur = R4[w4-i], prev = R4[w4-i-1].
    float s0, s1, s2, s3;
    if (DS == 0) { s0 = cur.x;  s1 = cur.y;  s2 = cur.z;  s3 = cur.w; }
    if (DS == 1) { s0 = prev.w; s1 = cur.x;  s2 = cur.y;  s3 = cur.z; }
    if (DS == 2) { s0 = prev.z; s1 = prev.w; s2 = cur.x;  s3 = cur.y; }
    if (DS == 3) { s0 = prev.y; s1 = prev.z; s2 = prev.w; s3 = cur.x; }
    // Mask: w0+k >= d  <=>  source index >= 0, so clamped loads never leak.
    v4f t;
    t.x = (w0 + 0 >= d) ? s0 : 0.0f;
    t.y = (w0 + 1 >= d) ? s1 : 0.0f;
    t.z = (w0 + 2 >= d) ? s2 : 0.0f;
    t.w = (w0 + 3 >= d) ? s3 : 0.0f;
    __builtin_nontemporal_store(t, reinterpret_cast<v4f*>(o + i * 4 * kHW));
    if (i < kDI - 1) {  // roll window one float4 down for d += 4
      cur = prev;
      int pidx = w0 - 8 - 4 * i;
      pidx = pidx > 0 ? pidx : 0;   // 16B-aligned clamp; masked anyway
      prev = *reinterpret_cast<const v4f*>(rrow + pidx);
    }
  }
}

__global__ __launch_bounds__(256) void concat_volume_kernel(
    const float* __restrict__ left, const float* __restrict__ right,
    float* __restrict__ out) {
  using namespace cv;
  // Block-uniform decomposition: pure SALU (all powers of two).
  const int blk  = blockIdx.x;          // 2 * B*C*H = 16384 blocks
  const int half = blk >> 13;           // 0 = left, 1 = right
  const int r    = blk & 8191;
  const int h    = r & (kH - 1);
  const int c    = (r >> 7) & (kC - 1);
  const int b    = r >> 12;

  const int w4   = threadIdx.x & 63;    // valid < 60
  const int dsub = threadIdx.x >> 6;    // 0..3 (wave-uniform)
  if (w4 >= kW4) return;
  const int w0 = w4 * 4;

  const float* srow = (half == 0 ? left : right) + ((b * kC + c) * kH + h) * kW;
  // out index: (((b*64 + c2)*48 + d)*128 + h)*240 + w,  c2 = c + half*32
  float* o = out + (((b * 64 + half * kC + c) * kD + dsub) * kH + h) * kW + w0;

  if (half == 0) {
    // ---- left half: broadcast one float4 across all d ----
    const v4f v = *reinterpret_cast<const v4f*>(srow + w0);
    if (w0 >= kD) {
      // Never masked (d <= 47 < w0): bare NT store loop.
#pragma unroll
      for (int i = 0; i < kDI; ++i)
        __builtin_nontemporal_store(v, reinterpret_cast<v4f*>(o + i * 4 * kHW));
    } else {
#pragma unroll
      for (int i = 0; i < kDI; ++i) {
        const int d = dsub + 4 * i;
        v4f t;
        t.x = (w0 + 0 >= d) ? v.x : 0.0f;
        t.y = (w0 + 1 >= d) ? v.y : 0.0f;
        t.z = (w0 + 2 >= d) ? v.z : 0.0f;
        t.w = (w0 + 3 >= d) ? v.w : 0.0f;
        __builtin_nontemporal_store(t, reinterpret_cast<v4f*>(o + i * 4 * kHW));
      }
    }
  } else {
    // ---- right half: d-shifted rolling-window gather ----
    const v4f cur  = *reinterpret_cast<const v4f*>(srow + w0);
    const int p0   = (w0 > 4) ? (w0 - 4) : 0;
    const v4f prev = *reinterpret_cast<const v4f*>(srow + p0);
    switch (dsub) {  // wave-uniform branch
      case 0:  right_loop<0>(srow, o, w0, cur, prev); break;
      case 1:  right_loop<1>(srow, o, w0, cur, prev); break;
      case 2:  right_loop<2>(srow, o, w0, cur, prev); break;
      default: right_loop<3>(srow, o, w0, cur, prev); break;
    }
  }
}

extern "C" void kernel_launch(void* const* d_in, const int* in_sizes, int n_in,
                              void* d_out, int out_size, void* d_ws, size_t ws_size,
                              hipStream_t stream) {
  (void)in_sizes; (void)n_in; (void)d_ws; (void)ws_size; (void)out_size;
  const float* left  = (const float*)d_in[0];
  const float* right = (const float*)d_in[1];
  // d_in[2] = max_disp (==48), compile-time constant here.
  float* out = (float*)d_out;

  constexpr int kThreads = 256;                            // 8 wave32 / block
  constexpr int kBlocks  = 2 * cv::kB * cv::kC * cv::kH;   // 16384
  concat_volume_kernel<<<kBlocks, kThreads, 0, stream>>>(left, right, out);
}